// FastCPLoss_83476984365677
// MI455X (gfx1250) — compile-verified
//
#include <hip/hip_runtime.h>
#include <stdint.h>

// Problem constants from the reference: N=2, C=150, H=W=512, ignore=255.
#define NCLS        150
#define HW_SHIFT    18              // H*W = 512*512 = 1<<18
#define HW          (1 << HW_SHIFT)
#define IGNORE_IDX  255
#define BLOCK       256             // 8 wave32s
#define GRID        512

typedef float v2f __attribute__((ext_vector_type(2)));
typedef float v8f __attribute__((ext_vector_type(8)));

__global__ void cp_init(float* acc) {
  acc[0] = 0.f;   // total loss
  acc[1] = 0.f;   // valid count
}

// Wave32 all-lanes sum via V_WMMA_F32_16X16X4_F32 with an all-ones B matrix.
// Each lane writes its partial into both A VGPRs: the 64 A-matrix slots are a
// bijection of the 64 (lane,vgpr) register slots, so sum(A) = 2 * sum(partials).
// With B==1: D[m,n] = rowsum_m(A). Per the documented C/D layout, lane n holds
// D[r, n] (r=0..7) and lane n+16 holds D[8+r, n], so summing a lane's 8 D regs
// and adding the xor-16 partner gives sum_m rowsum_m = sum(A).
__device__ __forceinline__ float wave_sum_wmma(float partial) {
  v2f a; a[0] = partial; a[1] = partial;
  v2f b; b[0] = 1.f;     b[1] = 1.f;
  v8f c = {0.f, 0.f, 0.f, 0.f, 0.f, 0.f, 0.f, 0.f};
  v8f d = __builtin_amdgcn_wmma_f32_16x16x4_f32(
      /*neg_a=*/false, a, /*neg_b=*/false, b,
      /*c_mod=*/(short)0, c, /*reuse_a=*/false, /*reuse_b=*/false);
  float h = ((d[0] + d[1]) + (d[2] + d[3])) + ((d[4] + d[5]) + (d[6] + d[7]));
  h += __shfl_xor(h, 16, 32);
  return 0.5f * h;
}

__global__ __launch_bounds__(BLOCK)
void cp_main(const float* __restrict__ logits,
             const int*   __restrict__ target,
             const float* __restrict__ S,
             float*       __restrict__ acc,
             int npix) {
  __shared__ float Dlds[NCLS * NCLS];        // 90 KB: penalty matrix D
  __shared__ float redL[BLOCK / 32];
  __shared__ float redC[BLOCK / 32];

  const int tid = threadIdx.x;

  // ---- Stage 1: async-copy S (150x150 f32 = 5625 x b128) into LDS ----
  {
    const unsigned long long sbase = (unsigned long long)S;
    const int nvec = (NCLS * NCLS) / 4;      // 5625
    for (int i = tid; i < nvec; i += BLOCK) {
      unsigned lds_addr = (unsigned)(uintptr_t)(&Dlds[i * 4]); // low 32b = LDS offset
      unsigned goff = (unsigned)(i * 16);
      asm volatile("global_load_async_to_lds_b128 %0, %1, %2"
                   :: "v"(lds_addr), "v"(goff), "s"(sbase)
                   : "memory");
    }
    asm volatile("s_wait_asynccnt 0x0" ::: "memory");
  }
  __syncthreads();

  // ---- Stage 2: transform S -> D = 1 - max(clip(S,0,1), I) in place ----
  for (int i = tid; i < NCLS * NCLS; i += BLOCK) {
    int y = i / NCLS;
    int cc = i - y * NCLS;
    float v = Dlds[i];
    v = fminf(fmaxf(v, 0.f), 1.f);
    if (y == cc) v = 1.f;
    Dlds[i] = 1.f - v;
  }
  __syncthreads();

  // ---- Stage 3: single pass over logits (memory-bound: 315 MB @ 23.3 TB/s) ----
  float lossAcc = 0.f, cntAcc = 0.f;
  const int stride = GRID * BLOCK;
  for (int pix = blockIdx.x * BLOCK + tid; pix < npix; pix += stride) {
    const int n = pix >> HW_SHIFT;
    const int p = pix & (HW - 1);
    const float* lp = logits + (size_t)n * (size_t)(NCLS * HW) + p;
    const int t = target[pix];
    const bool valid = (t != IGNORE_IDX);
    const float* __restrict__ drow = &Dlds[(valid ? t : 0) * NCLS];
    float s = 0.f, d = 0.f;
#pragma unroll 5
    for (int c = 0; c < NCLS; ++c) {
      float x = lp[(size_t)c << HW_SHIFT];   // coalesced 128B/wave per class plane
      x = fminf(x, 80.f);                    // overflow guard; softmax-equivalent
      float e = __expf(x);
      s += e;
      d = fmaf(e, drow[c], d);
    }
    if (valid) { lossAcc += d / s; cntAcc += 1.f; }
  }

  // ---- Stage 4: wave reduce (WMMA) -> block reduce (LDS) -> global atomics ----
  float waveLoss = wave_sum_wmma(lossAcc);
  float waveCnt  = wave_sum_wmma(cntAcc);

  const int lane = tid & 31, wid = tid >> 5;
  if (lane == 0) { redL[wid] = waveLoss; redC[wid] = waveCnt; }
  __syncthreads();
  if (tid == 0) {
    float L = 0.f, Cn = 0.f;
#pragma unroll
    for (int i = 0; i < BLOCK / 32; ++i) { L += redL[i]; Cn += redC[i]; }
    atomicAdd(&acc[0], L);
    atomicAdd(&acc[1], Cn);
  }
}

__global__ void cp_final(const float* acc, float* out) {
  float nv = acc[1] < 1.f ? 1.f : acc[1];
  out[0] = acc[0] / nv;
}

extern "C" void kernel_launch(void* const* d_in, const int* in_sizes, int n_in,
                              void* d_out, int out_size, void* d_ws, size_t ws_size,
                              hipStream_t stream) {
  const float* logits = (const float*)d_in[0];
  const int*   target = (const int*)d_in[1];   // integer input per harness convention
  const float* S      = (const float*)d_in[2];
  float* acc = (float*)d_ws;                   // [0]=total, [1]=valid count
  const int npix = in_sizes[1];                // N*H*W = 524288

  cp_init <<<1, 1, 0, stream>>>(acc);
  cp_main <<<GRID, BLOCK, 0, stream>>>(logits, target, S, acc, npix);
  cp_final<<<1, 1, 0, stream>>>(acc, (float*)d_out);
}